// TorStep_91001767067974
// MI455X (gfx1250) — compile-verified
//
#include <hip/hip_runtime.h>
#include <hip/hip_bf16.h>
#include <math.h>

// PET fused projection + backprojection for MI455X (gfx1250, wave32).
// Views: 0 = xlors, 1 = ylors, 2 = zlors. All three scatter into one
// accumulator (d_out), since the final result only needs their sum.
//
// Index permutations (vx,vy,vz = voxel coords in ray/pts space):
//   x-view: flat = vx*16384 + vz*128 + vy
//   y-view: flat = vz*16384 + vx*128 + vy
//   z-view: flat = vz*16384 + vy*128 + vx
// (identical for gather and scatter; verified against the reference's
//  transpose chain imgx/imgy/imgz + bpxt/bpyt/bpz.)

#define NSAMP 64
#define GN    128

__device__ __forceinline__ void atomic_add_f32_agent(float* p, float v) {
  // Non-returning fp32 global atomic add (L2-side), relaxed, device scope.
  __hip_atomic_fetch_add(p, v, __ATOMIC_RELAXED, __HIP_MEMORY_SCOPE_AGENT);
}

__device__ __forceinline__ int clamp127(int x) {
  return x < 0 ? 0 : (x > (GN - 1) ? (GN - 1) : x);
}

__global__ __launch_bounds__(256) void pet_fused_kernel(
    const float* __restrict__ img,
    const float* __restrict__ lx,
    const float* __restrict__ ly,
    const float* __restrict__ lz,
    float* __restrict__ accum,
    int nlors)
{
  const int t = blockIdx.x * blockDim.x + threadIdx.x;
  const int view = blockIdx.y;  // uniform per block -> scalar branches

  const float* __restrict__ lors = (view == 0) ? lx : ((view == 1) ? ly : lz);
  int mx, my, mz;
  if (view == 0)      { mx = 16384; my = 1;   mz = 128;   }  // x-view
  else if (view == 1) { mx = 128;   my = 1;   mz = 16384; }  // y-view
  else                { mx = 1;     my = 128; mz = 16384; }  // z-view

  if (t >= nlors) return;

  // lors stored as [7][nlors] row-major: coalesced per-component loads.
  const float p1x = lors[0 * nlors + t];
  const float p1y = lors[1 * nlors + t];
  const float p1z = lors[2 * nlors + t];
  const float dx  = lors[3 * nlors + t] - p1x;
  const float dy  = lors[4 * nlors + t] - p1y;
  const float dz  = lors[5 * nlors + t] - p1z;
  const float tc  = lors[6 * nlors + t];            // TOF_BIN = 1.0
  const float L   = sqrtf(dx * dx + dy * dy + dz * dz);
  const float wl  = (L * (1.0f / NSAMP)) * 3.0f;    // (L/64)*KERNEL_WIDTH

  // ---------------- Pass 1: projection (gather) ----------------
  float p = 0.0f;
#pragma unroll 1
  for (int s = 0; s < NSAMP; ++s) {
    const float u  = ((float)s + 0.5f) * (1.0f / NSAMP);
    const float qx = p1x + u * dx;
    const float qy = p1y + u * dy;
    const float qz = p1z + u * dz;
    const float dd = (u - 0.5f) * L - tc;
    const float w  = expf(dd * dd * (-0.5f / 100.0f)) * wl;  // TOF_SIGMA2=100

    const float vxr = (qx + 100.0f) * 0.64f - 0.5f;  // (pt - lo)/(size/grid) - 0.5
    const float vyr = (qy + 100.0f) * 0.64f - 0.5f;
    const float vzr = (qz + 100.0f) * 0.64f - 0.5f;
    const int ix = (int)floorf(vxr);
    const int iy = (int)floorf(vyr);
    const int iz = (int)floorf(vzr);
    const float fx = vxr - (float)ix;
    const float fy = vyr - (float)iy;
    const float fz = vzr - (float)iz;

    // Prefetch the voxel neighborhood 2 samples ahead (gfx1250
    // global_prefetch_b8); the ray advances predictably, so this hides
    // L2 latency on the otherwise-divergent gathers.
    {
      const float un = ((float)s + 2.5f) * (1.0f / NSAMP);
      const int px_ = clamp127((int)floorf((p1x + un * dx + 100.0f) * 0.64f - 0.5f));
      const int py_ = clamp127((int)floorf((p1y + un * dy + 100.0f) * 0.64f - 0.5f));
      const int pz_ = clamp127((int)floorf((p1z + un * dz + 100.0f) * 0.64f - 0.5f));
      __builtin_prefetch(img + (px_ * mx + py_ * my + pz_ * mz), 0, 3);
    }

    float val = 0.0f;
#pragma unroll
    for (int ox = 0; ox < 2; ++ox) {
      const int   jx = ix + ox;
      const float wx = ox ? fx : (1.0f - fx);
#pragma unroll
      for (int oy = 0; oy < 2; ++oy) {
        const int   jy = iy + oy;
        const float wy = oy ? fy : (1.0f - fy);
#pragma unroll
        for (int oz = 0; oz < 2; ++oz) {
          const int   jz = iz + oz;
          const float wz = oz ? fz : (1.0f - fz);
          if ((unsigned)jx < (unsigned)GN && (unsigned)jy < (unsigned)GN &&
              (unsigned)jz < (unsigned)GN) {
            const float cw = wx * wy * wz;
            val = fmaf(cw, img[jx * mx + jy * my + jz * mz], val);
          }
        }
      }
    }
    p = fmaf(val, w, p);
  }

  // ---------------- Pass 2: backprojection (scatter) ----------------
#pragma unroll 1
  for (int s = 0; s < NSAMP; ++s) {
    const float u  = ((float)s + 0.5f) * (1.0f / NSAMP);
    const float qx = p1x + u * dx;
    const float qy = p1y + u * dy;
    const float qz = p1z + u * dz;
    const float dd = (u - 0.5f) * L - tc;
    const float w  = expf(dd * dd * (-0.5f / 100.0f)) * wl;
    const float base = p * w;

    const float vxr = (qx + 100.0f) * 0.64f - 0.5f;
    const float vyr = (qy + 100.0f) * 0.64f - 0.5f;
    const float vzr = (qz + 100.0f) * 0.64f - 0.5f;
    const int ix = (int)floorf(vxr);
    const int iy = (int)floorf(vyr);
    const int iz = (int)floorf(vzr);
    const float fx = vxr - (float)ix;
    const float fy = vyr - (float)iy;
    const float fz = vzr - (float)iz;

#pragma unroll
    for (int ox = 0; ox < 2; ++ox) {
      const int   jx = ix + ox;
      const float wx = ox ? fx : (1.0f - fx);
#pragma unroll
      for (int oy = 0; oy < 2; ++oy) {
        const int   jy = iy + oy;
        const float wy = oy ? fy : (1.0f - fy);
#pragma unroll
        for (int oz = 0; oz < 2; ++oz) {
          const int   jz = iz + oz;
          const float wz = oz ? fz : (1.0f - fz);
          if ((unsigned)jx < (unsigned)GN && (unsigned)jy < (unsigned)GN &&
              (unsigned)jz < (unsigned)GN) {
            atomic_add_f32_agent(&accum[jx * mx + jy * my + jz * mz],
                                 base * (wx * wy * wz));
          }
        }
      }
    }
  }
}

__global__ __launch_bounds__(256) void zero_kernel(float* __restrict__ p, int n) {
  const int v = blockIdx.x * blockDim.x + threadIdx.x;
  if (v < n) p[v] = 0.0f;
}

__global__ __launch_bounds__(256) void finalize_kernel(
    const float* __restrict__ img,
    const float* __restrict__ eff,
    float* __restrict__ out, int n)
{
  const int v = blockIdx.x * blockDim.x + threadIdx.x;
  if (v < n) {
    const float a = out[v];              // accumulated bpx+bpy+bpz
    out[v] = img[v] / eff[v] * a;        // image/eff * sum  (canonical layout)
  }
}

extern "C" void kernel_launch(void* const* d_in, const int* in_sizes, int n_in,
                              void* d_out, int out_size, void* d_ws, size_t ws_size,
                              hipStream_t stream) {
  (void)n_in; (void)d_ws; (void)ws_size;
  const float* img = (const float*)d_in[0];
  const float* eff = (const float*)d_in[1];
  const float* lx  = (const float*)d_in[2];
  const float* ly  = (const float*)d_in[3];
  const float* lz  = (const float*)d_in[4];
  float* out = (float*)d_out;

  const int nlors = in_sizes[2] / 7;       // lors stored as [7][N]
  const int nvox  = out_size;              // 128^3

  const int TB = 256;
  // d_out doubles as the shared backprojection accumulator: zero it first.
  zero_kernel<<<(nvox + TB - 1) / TB, TB, 0, stream>>>(out, nvox);

  dim3 grid((nlors + TB - 1) / TB, 3);     // y-dim = view (x, y, z LOR sets)
  pet_fused_kernel<<<grid, TB, 0, stream>>>(img, lx, ly, lz, out, nlors);

  finalize_kernel<<<(nvox + TB - 1) / TB, TB, 0, stream>>>(img, eff, out, nvox);
}